// ConvWithFFT_10746008175004
// MI455X (gfx1250) — compile-verified
//
#include <hip/hip_runtime.h>
#include <math.h>

// ---------------------------------------------------------------------------
// Convolution via FFT on MI455X (gfx1250).
//   y = real( ifft( fft(pad(a)) * fft(pad(x)) ) )[:N],  N = 2^23, M = 2N = 2^24
//
// Bandwidth-bound pipeline (~3.8 GB traffic -> ~165us at 23.3 TB/s):
//   * pack z = a + i*x (real-packing trick) -> ONE forward FFT instead of two
//   * radix-16 Stockham autosort FFT: 6 passes for 2^24 points
//   * butterflies 16-at-a-time as a 16x16 complex matmul with fp32 WMMA
//     (V_WMMA_F32_16X16X4_F32; f16/bf16 would wreck a 16M-pt transform)
//   * twiddles via 5 sincos/tile + exact (-i)^t / step-4 complex recurrences
//     to keep the stage kernel memory-bound, not trans-bound
//   * stage 0 fully specialized: no twiddle multiply at all (tw == 1)
//   * spectral kernel separates Fa/Fx, multiplies, writes conj(Y) so the
//     inverse FFT is just another forward FFT; finalize scales Re by 1/M
// Workspace: 2 ping-pong complex buffers of 2^24 float2 = 256 MiB in d_ws.
// ---------------------------------------------------------------------------

typedef __attribute__((ext_vector_type(2))) float v2f;
typedef __attribute__((ext_vector_type(8))) float v8f;

#define FFT_LOG2M 24
#define FFT_M     (1 << FFT_LOG2M)     // 16777216
#define PI_F      3.14159265358979323846f

__device__ __forceinline__ v8f wmma_f32_16x16x4(v2f a, v2f b, v8f c) {
  // 8 args: (neg_a, A, neg_b, B, c_mod, C, reuse_a, reuse_b)
  return __builtin_amdgcn_wmma_f32_16x16x4_f32(
      false, a, false, b, (short)0, c, false, false);
}

// z[2i],z[2i+1] = (a,x) pairs for 2i<n, else 0 — b128 streaming writes.
__global__ __launch_bounds__(256) void pack_kernel(
    const float* __restrict__ a, const float* __restrict__ x,
    float4* __restrict__ z2, int n) {
  int i  = blockIdx.x * 256 + threadIdx.x;   // over M/2
  int i2 = i << 1;
  float4 v = make_float4(0.0f, 0.0f, 0.0f, 0.0f);
  if (i2 < n) { v.x = a[i2]; v.y = x[i2]; v.z = a[i2 + 1]; v.w = x[i2 + 1]; }
  z2[i] = v;
}

// One radix-16 Stockham stage (forward, ω = exp(-2πi/·)), Ns = 16^stage.
//   j = q*Ns + k ; dst[q*16Ns + t*Ns + k] = Σ_m ω16^{tm} ω_{16Ns}^{km} src[m*M/16 + j]
// One wave = 16 consecutive butterflies = Y(16x16) = W16 × (tw∘X):
// 4 real 16x16x16 matmuls = 16 accumulating K=4 fp32 WMMAs.
// Fragment layout (ISA): lane L -> row/col = L%16, K = v + 2*(L/16).
template <int LOG2NS>
__global__ __launch_bounds__(256) void fft_stage16(
    const float2* __restrict__ src, float2* __restrict__ dst) {
  constexpr int Mdiv16 = FFT_M >> 4;
  constexpr int Ns     = 1 << LOG2NS;

  const int lane = threadIdx.x & 31;
  const int wave = threadIdx.x >> 5;          // wave32
  const int tile = (blockIdx.x << 3) + wave;  // 8 waves per block
  const int j0   = tile << 4;

  const int col  = lane & 15;
  const int half = lane >> 4;
  const int j    = j0 + col;
  const int k    = j & (Ns - 1);
  const int q    = j >> LOG2NS;

  // The two K rows this lane feeds per chunk: m = m0 + mA / m0 + mB
  const int mA = half << 1;
  const int mB = mA + 1;

  // A operand: W16[t=col][m] = ω16^{t m}; base values for chunk 0.
  float a0r, a0i, a1r, a1i;
  {
    float s, c;
    __sincosf((-PI_F / 8.0f) * (float)((col * mA) & 15), &s, &c);
    a0r = c; a0i = s;
    __sincosf((-PI_F / 8.0f) * (float)((col * mB) & 15), &s, &c);
    a1r = c; a1i = s;
  }
  // Exact chunk rotation: ω16^{4t} = (-i)^t  ->  {(1,0),(0,-1),(-1,0),(0,1)}
  const int   tm = col & 3;
  const float rr = (tm == 0) ? 1.0f : ((tm == 2) ? -1.0f : 0.0f);
  const float ri = (tm == 1) ? -1.0f : ((tm == 3) ? 1.0f : 0.0f);

  // Stage twiddles tw(k,m) = ω_{16Ns}^{k m}; base values + step tw(k,4).
  float t0r = 1.0f, t0i = 0.0f, t1r = 1.0f, t1i = 0.0f;
  float s4r = 1.0f, s4i = 0.0f;
  if constexpr (LOG2NS > 0) {
    constexpr int   P = Ns << 4;
    constexpr float u = -2.0f * PI_F / (float)P;
    float s, c;
    __sincosf(u * (float)((k * mA) & (P - 1)), &s, &c); t0r = c; t0i = s;
    __sincosf(u * (float)((k * mB) & (P - 1)), &s, &c); t1r = c; t1i = s;
    __sincosf(u * (float)((k * 4) & (P - 1)), &s, &c);  s4r = c; s4i = s;
  }

  v8f acc_re = {};
  v8f acc_im = {};

#pragma unroll
  for (int m0 = 0; m0 < 16; m0 += 4) {
    // b64 loads; lanes 0-15 / 16-31 each form a contiguous 128B segment
    const float2 x0 = src[(m0 + mA) * Mdiv16 + j];
    const float2 x1 = src[(m0 + mB) * Mdiv16 + j];

    // B = tw * x   (stage 0: tw == 1, no FP ops at all)
    v2f Bre, Bim;
    if constexpr (LOG2NS == 0) {
      Bre.x = x0.x;  Bim.x = x0.y;
      Bre.y = x1.x;  Bim.y = x1.y;
    } else {
      Bre.x = x0.x * t0r - x0.y * t0i;  Bim.x = x0.x * t0i + x0.y * t0r;
      Bre.y = x1.x * t1r - x1.y * t1i;  Bim.y = x1.x * t1i + x1.y * t1r;
    }

    v2f Are, Aim, AnegIm;
    Are.x = a0r;  Are.y = a1r;
    Aim.x = a0i;  Aim.y = a1i;
    AnegIm.x = -a0i;  AnegIm.y = -a1i;

    // Yre += Wre·Bre - Wim·Bim ; Yim += Wre·Bim + Wim·Bre
    acc_re = wmma_f32_16x16x4(Are,    Bre, acc_re);
    acc_re = wmma_f32_16x16x4(AnegIm, Bim, acc_re);
    acc_im = wmma_f32_16x16x4(Are,    Bim, acc_im);
    acc_im = wmma_f32_16x16x4(Aim,    Bre, acc_im);

    if (m0 < 12) {
      float nr;
      // A *= (-i)^t  (exact rotation, no trans ops)
      nr = a0r * rr - a0i * ri;  a0i = a0r * ri + a0i * rr;  a0r = nr;
      nr = a1r * rr - a1i * ri;  a1i = a1r * ri + a1i * rr;  a1r = nr;
      if constexpr (LOG2NS > 0) {
        // tw *= tw(k,4)
        nr = t0r * s4r - t0i * s4i;  t0i = t0r * s4i + t0i * s4r;  t0r = nr;
        nr = t1r * s4r - t1i * s4i;  t1i = t1r * s4i + t1i * s4r;  t1r = nr;
      }
    }
  }

  // D layout: lane holds rows t = vv + 8*half of its column's butterfly
  const int obase = (q << (LOG2NS + 4)) + k;
#pragma unroll
  for (int vv = 0; vv < 8; ++vv) {
    const int t = vv + (half << 3);
    dst[obase + (t << LOG2NS)] = make_float2(acc_re[vv], acc_im[vv]);
  }
}

// From Z = fft(a + i·x): Fa = (Z[k]+conj(Z[M-k]))/2, Fx = (Z[k]-conj(Z[M-k]))/(2i)
// Y = Fa*Fx (Hermitian since y real). Write V = conj(Y):
//   V[k] = conj(Y[k]),  V[M-k] = conj(Y[M-k]) = Y[k].
__global__ __launch_bounds__(256) void spectral_kernel(
    const float2* __restrict__ Z, float2* __restrict__ V) {
  const int M = FFT_M;
  int k = blockIdx.x * 256 + threadIdx.x;
  if (k > (M >> 1)) return;

  const float2 zk = Z[k];
  const float2 zm = Z[(M - k) & (M - 1)];

  const float far = 0.5f * (zk.x + zm.x);
  const float fai = 0.5f * (zk.y - zm.y);
  const float fxr = 0.5f * (zk.y + zm.y);
  const float fxi = -0.5f * (zk.x - zm.x);

  const float yr = far * fxr - fai * fxi;
  const float yi = far * fxi + fai * fxr;

  V[k] = make_float2(yr, -yi);
  if (k != 0 && k != (M >> 1)) V[M - k] = make_float2(yr, yi);
}

// ifft(Y) = conj(fft(conj(Y)))/M -> out = Re(fft(V))/M ; b128 reads, b64 writes
__global__ __launch_bounds__(256) void finalize_kernel(
    const float4* __restrict__ G2, float2* __restrict__ out2, float scale) {
  int i = blockIdx.x * 256 + threadIdx.x;   // over N/2
  const float4 g = G2[i];
  out2[i] = make_float2(g.x * scale, g.z * scale);
}

static void launch_stage(const float2* s, float2* d, int st, hipStream_t stream) {
  const int blocks = (FFT_M / 256) / 8;   // one 16-butterfly tile per wave
  switch (st) {
    case 0: fft_stage16<0> <<<blocks, 256, 0, stream>>>(s, d); break;
    case 1: fft_stage16<4> <<<blocks, 256, 0, stream>>>(s, d); break;
    case 2: fft_stage16<8> <<<blocks, 256, 0, stream>>>(s, d); break;
    case 3: fft_stage16<12><<<blocks, 256, 0, stream>>>(s, d); break;
    case 4: fft_stage16<16><<<blocks, 256, 0, stream>>>(s, d); break;
    default: fft_stage16<20><<<blocks, 256, 0, stream>>>(s, d); break;
  }
}

extern "C" void kernel_launch(void* const* d_in, const int* in_sizes, int n_in,
                              void* d_out, int out_size, void* d_ws, size_t ws_size,
                              hipStream_t stream) {
  const float* a = (const float*)d_in[0];
  const float* x = (const float*)d_in[1];
  const int n = in_sizes[0];  // 8388608 = FFT_M/2

  float2* bufA = (float2*)d_ws;        // 128 MiB
  float2* bufB = bufA + FFT_M;         // 128 MiB (needs 256 MiB of d_ws)

  // 1) pack z = a + i·x, zero-padded to M
  pack_kernel<<<(FFT_M / 2) / 256, 256, 0, stream>>>(a, x, (float4*)bufA, n);

  // 2) forward FFT: 6 radix-16 stages, ping-pong A<->B, result lands in bufA
  {
    float2* s = bufA;
    float2* d = bufB;
    for (int st = 0; st < 6; ++st) {
      launch_stage(s, d, st, stream);
      float2* t = s; s = d; d = t;
    }
  }

  // 3) spectral separation + product; writes conj(Y) into bufB
  spectral_kernel<<<((FFT_M / 2 + 1) + 255) / 256, 256, 0, stream>>>(bufA, bufB);

  // 4) inverse FFT as forward FFT of conj(Y): 6 stages B<->A, result in bufB
  {
    float2* s = bufB;
    float2* d = bufA;
    for (int st = 0; st < 6; ++st) {
      launch_stage(s, d, st, stream);
      float2* t = s; s = d; d = t;
    }
  }

  // 5) out = Re(result)/M, first N elements
  finalize_kernel<<<(n / 2) / 256, 256, 0, stream>>>(
      (const float4*)bufB, (float2*)d_out, 1.0f / (float)FFT_M);
}